// LocallyConnected2d_66425964200603
// MI455X (gfx1250) — compile-verified
//
#include <hip/hip_runtime.h>

// LocallyConnected2d: out[b,o,h,w] = sum_{c,k} patch[b,c,h,w,k] * weight[o,c,h,w,k] + bias
// Per-location GEMM (B=8 x CK=144) x (CK=144 x O=32) via V_WMMA_F32_16X16X4_F32.
// LDS operands stored in WMMA *fragment order* so inner-loop DS loads use pure
// immediate offsets (no per-step VALU address math).

#define B_   8
#define C_   16
#define H_   64
#define W_   64
#define O_   32
#define K2_  9
#define HO_  62
#define WO_  62
#define CCH  4                         // c values per LDS chunk
#define NCH  (C_ / CCH)                // 4 chunks
#define NT   9                         // WMMA K-steps per chunk (36 ck / 4)
#define LPB  8                         // locations (w) per block, one per wave
#define WBLK ((WO_ + LPB - 1) / LPB)   // 8
#define WCHUNK (O_ * CCH * LPB * K2_)  // 9216 weight floats per chunk
#define ACHUNK (18 * LPB * B_ * 2)     // 2304 A floats per chunk (18 pairs)

// fragment-order LDS strides (odd pair strides -> halves on opposite bank parity)
#define PW   521                       // lwT stride per ck-pair P
#define WLW  66                        // lwT stride per local w
#define LWT_SZ (17 * PW + 7 * WLW + 31 * 2 + 1 + 1)   // 9383 floats
#define PA   131                       // lxT stride per ck-pair P
#define WLA  16                        // lxT stride per local w
#define LXT_SZ (17 * PA + 7 * WLA + 7 * 2 + 1 + 1)    // 2355 floats

typedef __attribute__((ext_vector_type(2))) float v2f;
typedef __attribute__((ext_vector_type(8))) float v8f;

__global__ __launch_bounds__(256, 1)
void lc2d_wmma_f32_kernel(const float* __restrict__ x,
                          const float* __restrict__ wgt,
                          const float* __restrict__ bias,
                          float* __restrict__ out)
{
    __shared__ float lwT[LWT_SZ];   // weights, fragment order: [P][wl][o][v]
    __shared__ float lxT[LXT_SZ];   // patches, fragment order: [P][wl][b][v]

    const int tid  = threadIdx.x;
    const int lane = tid & 31;
    const int wave = tid >> 5;              // local w
    const int bid  = blockIdx.x;
    const int h    = bid / WBLK;
    const int w0   = (bid % WBLK) * LPB;
    const int w    = w0 + wave;
    const bool active = (w < WO_);          // wave-uniform -> EXEC all-ones inside

    const int half = lane >> 4;             // 0: K pair {4t,4t+1}, 1: {4t+2,4t+3}
    const int m    = lane & 7;              // batch row (rows 8-15 duplicate 0-7)
    const int n    = lane & 15;             // o within tile

    // per-lane fragment base indices (all inner-loop offsets are immediates)
    const int lwBase = half * PW + wave * WLW + n * 2;
    const int lxBase = half * PA + wave * WLA + m * 2;

    v8f acc0 = {};                          // o = 0..15
    v8f acc1 = {};                          // o = 16..31

    // valid flat (wl*9+k) extent of this block's contiguous weight segments
    const int rmax = (w0 + LPB <= WO_) ? (LPB * K2_) : ((WO_ - w0) * K2_);

    for (int cc = 0; cc < NCH; ++cc) {
        __syncthreads();

        // ---- stage weight chunk: coalesced 72-float segments -> fragment order ----
        for (int i = tid; i < WCHUNK; i += 256) {
            int p   = i / (LPB * K2_);      // (o, cl)
            int r   = i % (LPB * K2_);      // wl*9 + k : contiguous in memory
            int o   = p >> 2;
            int cl  = p & 3;
            int wl  = r / K2_;
            int k   = r % K2_;
            int ckl = cl * K2_ + k;         // 0..35 within chunk
            float v = 0.0f;
            if (r < rmax)
                v = wgt[(((o * C_ + (cc * CCH + cl)) * HO_ + h) * WO_ + w0) * K2_ + r];
            lwT[(ckl >> 1) * PW + wl * WLW + o * 2 + (ckl & 1)] = v;
        }

        // ---- stage A chunk: x patches -> fragment order (hot 15KB region) ----
        for (int i = tid; i < ACHUNK; i += 256) {
            int P   = i >> 7;               // ck pair 0..17
            int r   = i & 127;
            int wl  = r >> 4;
            int b   = (r >> 1) & 7;
            int v   = r & 1;
            int ckl = 2 * P + v;
            int cg  = cc * CCH + ckl / K2_;
            int k   = ckl % K2_;
            int ww  = w0 + wl + (k % 3);
            float val = 0.0f;
            if (ww < W_)
                val = x[((b * C_ + cg) * H_ + (h + k / 3)) * W_ + ww];
            lxT[P * PA + wl * WLA + b * 2 + v] = val;
        }

        // prefetch next weight chunk (one cacheline per 32 elems) behind the WMMAs
        if (cc + 1 < NCH) {
            for (int i = tid * 32; i < WCHUNK; i += 256 * 32) {
                int p  = i / (LPB * K2_);
                int r  = i % (LPB * K2_);
                int o  = p >> 2;
                int cl = p & 3;
                __builtin_prefetch(
                    &wgt[(((o * C_ + ((cc + 1) * CCH + cl)) * HO_ + h) * WO_ + w0) * K2_ + r],
                    0, 1);
            }
        }
        __syncthreads();

        if (active) {
#pragma unroll
            for (int t = 0; t < NT; ++t) {
                v2f a, b0, b1;
                a.x  = lxT[lxBase + t * (2 * PA)    ];
                a.y  = lxT[lxBase + t * (2 * PA) + 1];
                b0.x = lwT[lwBase + t * (2 * PW)    ];
                b0.y = lwT[lwBase + t * (2 * PW) + 1];
                b1.x = lwT[lwBase + t * (2 * PW) + 32];
                b1.y = lwT[lwBase + t * (2 * PW) + 33];

                acc0 = __builtin_amdgcn_wmma_f32_16x16x4_f32(
                           false, a, false, b0, (short)0, acc0, false, false);
                acc1 = __builtin_amdgcn_wmma_f32_16x16x4_f32(
                           false, a, false, b1, (short)0, acc1, false, false);
            }
        }
    }

    if (active) {
        // D layout: VGPR r, lanes 0-15 -> (b=r, o=lane) from acc0;
        // acc1 lanes 16-31 duplicate batch rows -> (b=r, o=lane) as well.
        const float bv = bias[(lane * HO_ + h) * WO_ + w];
#pragma unroll
        for (int r = 0; r < B_; ++r) {
            float v = (lane < 16) ? acc0[r] : acc1[r];
            out[((r * O_ + lane) * HO_ + h) * WO_ + w] = v + bv;
        }
    }
}

extern "C" void kernel_launch(void* const* d_in, const int* in_sizes, int n_in,
                              void* d_out, int out_size, void* d_ws, size_t ws_size,
                              hipStream_t stream) {
    const float* x    = (const float*)d_in[0];   // [8,16,64,64]
    const float* wgt  = (const float*)d_in[1];   // [32,16,62,62,9]
    const float* bias = (const float*)d_in[2];   // [1,32,62,62]
    float* out        = (float*)d_out;           // [8,32,62,62]

    dim3 grid(HO_ * WBLK);   // 62 rows * 8 w-chunks = 496 blocks
    dim3 block(256);         // 8 waves; one output location per wave
    hipLaunchKernelGGL(lc2d_wmma_f32_kernel, grid, block, 0, stream,
                       x, wgt, bias, out);
}